// VQVAE_70360154243133
// MI455X (gfx1250) — compile-verified
//
#include <hip/hip_runtime.h>

typedef __attribute__((ext_vector_type(2))) float v2f;
typedef __attribute__((ext_vector_type(8))) float v8f;

#define B_ 32
#define D_ 64
#define H_ 32
#define W_ 32
#define HW_ (H_ * W_)            // 1024
#define K_ 2048
#define NROWS (B_ * HW_)         // 32768
#define ROWS_PER_WG 128
#define NTILES (K_ / 16)         // 128 column tiles
#define TILE_FLOATS (16 * D_)    // 1024 floats between col tiles in codebook
#define PAIR_STRIDE 260          // dwords per d-pair row: 128 rows * 2 + 4 pad

// ---------------------------------------------------------------------------
// Kernel 1: esq_half[k] = 0.5 * sum_d codebook[k][d]^2
// ---------------------------------------------------------------------------
__global__ __launch_bounds__(256) void esq_kernel(const float* __restrict__ cb,
                                                  float* __restrict__ esq_half) {
    int k = blockIdx.x * 256 + threadIdx.x;
    const float4* row = (const float4*)(cb + (size_t)k * D_);
    float s = 0.f;
#pragma unroll
    for (int i = 0; i < D_ / 4; ++i) {
        float4 v = row[i];
        s += v.x * v.x + v.y * v.y + v.z * v.z + v.w * v.w;
    }
    esq_half[k] = 0.5f * s;
}

// ---- helpers -------------------------------------------------------------

// One clause of 16 global b64 loads into distinct registers (no wait needed
// until first use -> overlaps with the WMMA chain on the other buffer).
__device__ __forceinline__ void load_btile(v2f (&bf)[16], const float* brow) {
#pragma unroll
    for (int kk = 0; kk < 16; ++kk)
        bf[kk] = *(const v2f*)(brow + 4 * kk);
}

// 16-deep K accumulation split over two accumulators to break the
// back-to-back same-C WMMA dependency, then fused argmax update.
__device__ __forceinline__ void tile_update(const v2f (&afrag)[16], const v2f (&bf)[16],
                                            float eh, int col,
                                            float (&best)[8], int (&bidx)[8]) {
    v8f c0 = {}, c1 = {};
#pragma unroll
    for (int kk = 0; kk < 16; kk += 2) {
        c0 = __builtin_amdgcn_wmma_f32_16x16x4_f32(
                 false, afrag[kk],     false, bf[kk],     (short)0, c0, false, false);
        c1 = __builtin_amdgcn_wmma_f32_16x16x4_f32(
                 false, afrag[kk + 1], false, bf[kk + 1], (short)0, c1, false, false);
    }
#pragma unroll
    for (int j = 0; j < 8; ++j) {
        float s = (c0[j] + c1[j]) - eh;            // row M = j + 8*hi
        if (s > best[j]) { best[j] = s; bidx[j] = col; }   // strict > keeps first idx
    }
}

// ---------------------------------------------------------------------------
// Kernel 2: fused GEMM (WMMA f32 16x16x4) + argmax + gather.
//   score[n][k] = x_n . e_k - 0.5*||e_k||^2   (argmax == reference argmin)
// ---------------------------------------------------------------------------
__global__ __launch_bounds__(256) void vq_kernel(const float* __restrict__ laten,
                                                 const float* __restrict__ cb,
                                                 const float* __restrict__ esq_half,
                                                 float* __restrict__ out_idx,
                                                 float* __restrict__ out_q) {
    __shared__ float ldsX[32 * PAIR_STRIDE];   // 32 d-pairs x 128 rows (pair-interleaved)
    __shared__ int   ldsIdx[ROWS_PER_WG];

    const int t    = threadIdx.x;
    const int lane = t & 31;
    const int wave = t >> 5;

    const int row0 = blockIdx.x * ROWS_PER_WG;     // global row base
    const int b    = row0 >> 10;                   // row0 / HW (128 | 1024 -> one image)
    const int hw0  = row0 & (HW_ - 1);

    // ---- stage x tile into LDS, transposed to (d, row), d-pairs adjacent ----
    const float* src = laten + (size_t)b * D_ * HW_ + hw0;
#pragma unroll
    for (int j = 0; j < 32; ++j) {
        int i = t + 256 * j;          // 0..8191
        int d = i >> 7;               // 0..63
        int r = i & 127;              // 0..127
        ldsX[(d >> 1) * PAIR_STRIDE + r * 2 + (d & 1)] = src[(size_t)d * HW_ + r];
    }
    __syncthreads();

    // ---- A fragments (f32 16x16x4 layout), loaded once, reused 128 tiles ----
    // lane 0-15: (K=0,K=1); lane 16-31: (K=2,K=3).  d0 = 4*kk + 2*hi (even).
    const int hi   = lane >> 4;
    const int lrow = wave * 16 + (lane & 15);
    v2f afrag[16];
#pragma unroll
    for (int kk = 0; kk < 16; ++kk) {
        int d0 = 4 * kk + 2 * hi;
        afrag[kk] = *(const v2f*)&ldsX[(d0 >> 1) * PAIR_STRIDE + lrow * 2];
    }

    float best[8];
    int   bidx[8];
#pragma unroll
    for (int j = 0; j < 8; ++j) { best[j] = -3.4e38f; bidx[j] = 0; }

    // ---- column-tile sweep, register double-buffered (unrolled x2 so the
    //      buffer choice is compile-time: no dynamic VGPR indexing) ----
    const int ncol = lane & 15;
    const float* bbase = cb + (size_t)ncol * D_ + 2 * hi;  // B frag base, tile 0

    v2f  bf0[16], bf1[16];
    float eh0, eh1;
    load_btile(bf0, bbase);                 // prologue: tile 0 in flight
    eh0 = esq_half[ncol];

    for (int ct = 0; ct < NTILES; ct += 2) {
        load_btile(bf1, bbase + (size_t)(ct + 1) * TILE_FLOATS);   // tile ct+1 loads
        eh1 = esq_half[(ct + 1) * 16 + ncol];
        tile_update(afrag, bf0, eh0, ct * 16 + ncol, best, bidx);  // compute tile ct

        if (ct + 2 < NTILES) {
            load_btile(bf0, bbase + (size_t)(ct + 2) * TILE_FLOATS); // tile ct+2 loads
            eh0 = esq_half[(ct + 2) * 16 + ncol];
        }
        tile_update(afrag, bf1, eh1, (ct + 1) * 16 + ncol, best, bidx); // tile ct+1
    }

    // ---- half-wave argmax reduction (wave32: masks<16 stay in each half) ----
#pragma unroll
    for (int j = 0; j < 8; ++j) {
        float bv = best[j];
        int   bi = bidx[j];
#pragma unroll
        for (int m = 8; m >= 1; m >>= 1) {
            float ov = __shfl_xor(bv, m, 32);
            int   oi = __shfl_xor(bi, m, 32);
            if (ov > bv || (ov == bv && oi < bi)) { bv = ov; bi = oi; }
        }
        if ((lane & 15) == 0) {
            int rloc = wave * 16 + hi * 8 + j;      // row within WG tile
            ldsIdx[rloc]         = bi;
            out_idx[row0 + rloc] = (float)bi;
        }
    }
    __syncthreads();

    // ---- gather: out_q[b][d][hw0+r] = cb[idx[r]][d], coalesced along r ----
    float* dst = out_q + (size_t)b * D_ * HW_ + hw0;
#pragma unroll
    for (int j = 0; j < 32; ++j) {
        int i = t + 256 * j;
        int d = i >> 7;
        int r = i & 127;
        dst[(size_t)d * HW_ + r] = cb[(size_t)ldsIdx[r] * D_ + d];
    }
}

// ---------------------------------------------------------------------------
extern "C" void kernel_launch(void* const* d_in, const int* in_sizes, int n_in,
                              void* d_out, int out_size, void* d_ws, size_t ws_size,
                              hipStream_t stream) {
    const float* laten = (const float*)d_in[0];
    const float* cb    = (const float*)d_in[1];
    float* esq_half    = (float*)d_ws;             // 2048 floats = 8 KB scratch
    float* out         = (float*)d_out;            // [0,32768): idx, then quant_laten

    esq_kernel<<<K_ / 256, 256, 0, stream>>>(cb, esq_half);
    vq_kernel<<<NROWS / ROWS_PER_WG, 256, 0, stream>>>(laten, cb, esq_half,
                                                       out, out + NROWS);
}